// TopologicalLoss_74242804678735
// MI455X (gfx1250) — compile-verified
//
#include <hip/hip_runtime.h>
#include <math.h>

typedef __attribute__((ext_vector_type(2))) float v2f;
typedef __attribute__((ext_vector_type(8))) float v8f;

constexpr int N = 4096;
constexpr int K = 128;

// ---------------------------------------------------------------------------
// Squared norms of every row, both views (f32, full precision).
// ---------------------------------------------------------------------------
__global__ void __launch_bounds__(256)
sqnorm_kernel(const float* __restrict__ x1, const float* __restrict__ x2,
              float* __restrict__ sq1, float* __restrict__ sq2) {
    int t = blockIdx.x * blockDim.x + threadIdx.x;   // 0 .. 2N-1
    if (t >= 2 * N) return;
    const float* x = (t < N) ? x1 : x2;
    float* sq      = (t < N) ? sq1 : sq2;
    int row        = (t < N) ? t : t - N;
    const float4* p = (const float4*)(x + (size_t)row * K);
    float s = 0.f;
#pragma unroll
    for (int i = 0; i < K / 4; ++i) {
        float4 v = p[i];
        s += v.x * v.x + v.y * v.y + v.z * v.z + v.w * v.w;
    }
    sq[row] = s;
}

// ---------------------------------------------------------------------------
// Distance matrix via fp32 WMMA (V_WMMA_F32_16X16X4_F32).
// Each wave: one 16-row A strip cached in registers, 4 consecutive 16x16
// j-tiles (16x64 output strip).  D[i][j] = sqrt(max(|xi|^2+|xj|^2-2<xi,xj>,0)+1e-12)
// ---------------------------------------------------------------------------
__global__ void __launch_bounds__(128)
dist_wmma_kernel(const float* __restrict__ x1, const float* __restrict__ x2,
                 const float* __restrict__ sq1, const float* __restrict__ sq2,
                 float* __restrict__ D1, float* __restrict__ D2) {
    const int view      = blockIdx.z;
    const float* x      = view ? x2 : x1;
    const float* sq     = view ? sq2 : sq1;
    float* Dm           = view ? D2 : D1;

    const int wave = threadIdx.x >> 5;     // 4 waves / block (wave32)
    const int lane = threadIdx.x & 31;
    const int l    = lane & 15;
    const int hi   = lane >> 4;

    const int i0    = (blockIdx.y * 4 + wave) * 16;  // row tile
    const int jbase = blockIdx.x * 64;               // 4 col tiles

    // A operand: lane l holds row i0+l; per k-step s it holds K = 4s+2*hi, 4s+2*hi+1
    v2f a[K / 4];
    const float* arow = x + (size_t)(i0 + l) * K + 2 * hi;
#pragma unroll
    for (int s = 0; s < K / 4; ++s)
        a[s] = *(const v2f*)(arow + s * 4);

#pragma unroll
    for (int jt = 0; jt < 4; ++jt) {
        const int j0 = jbase + jt * 16;
        // B operand: lane l holds column j0+l (= x row j0+l), same K striping as A
        const float* brow = x + (size_t)(j0 + l) * K + 2 * hi;
        v8f c = {};
#pragma unroll
        for (int s = 0; s < K / 4; ++s) {
            v2f b = *(const v2f*)(brow + s * 4);
            c = __builtin_amdgcn_wmma_f32_16x16x4_f32(
                    /*neg_a=*/false, a[s], /*neg_b=*/false, b,
                    /*c_mod=*/(short)0, c, /*reuse_a=*/false, /*reuse_b=*/false);
        }
        const float sqc = sq[j0 + l];
#pragma unroll
        for (int r = 0; r < 8; ++r) {
            const int row = i0 + r + 8 * hi;          // C/D layout: M = r + 8*hi
            const float d2 = sq[row] + sqc - 2.0f * c[r];
            Dm[(size_t)row * N + (j0 + l)] = sqrtf(fmaxf(d2, 0.0f) + 1e-12f);
        }
    }
}

// ---------------------------------------------------------------------------
// Prim's MST, one workgroup per view.  dist[] lives in LDS; each iteration:
// argmin over non-tree nodes (tree reduction, first-index tie-break),
// then dist = min(dist, D[j]) streaming row j out of L2 (D fits in 192MB L2).
// Writes N-1 capped edge weights; slot N-1 padded with +inf for the sorter.
// ---------------------------------------------------------------------------
__global__ void __launch_bounds__(256)
prim_kernel(const float* __restrict__ D1, const float* __restrict__ D2,
            float* __restrict__ w1, float* __restrict__ w2) {
    const float* Dm = blockIdx.x ? D2 : D1;
    float* w        = blockIdx.x ? w2 : w1;
    const int tid = threadIdx.x;

    __shared__ float dist[N];
    __shared__ unsigned char flag[N];
    __shared__ float rv[256];
    __shared__ int   ri[256];

    for (int i = tid; i < N; i += 256) { dist[i] = Dm[i]; flag[i] = 0; }
    if (tid == 0) { flag[0] = 1; w[N - 1] = INFINITY; }
    __syncthreads();

    for (int it = 0; it < N - 1; ++it) {
        float best = INFINITY; int bidx = N;
        for (int i = tid; i < N; i += 256) {
            if (!flag[i]) {
                float v = dist[i];
                if (v < best) { best = v; bidx = i; }   // strided asc -> first idx kept
            }
        }
        rv[tid] = best; ri[tid] = bidx;
        __syncthreads();
        for (int s = 128; s > 0; s >>= 1) {
            if (tid < s) {
                float ov = rv[tid + s]; int oi = ri[tid + s];
                if (ov < rv[tid] || (ov == rv[tid] && oi < ri[tid])) {
                    rv[tid] = ov; ri[tid] = oi;
                }
            }
            __syncthreads();
        }
        const int j = ri[0];
        if (tid == 0) { w[it] = fminf(rv[0], 100.0f); flag[j] = 1; }
        __syncthreads();
        const float* row = Dm + (size_t)j * N;
        for (int i = tid; i < N; i += 256)
            if (!flag[i]) dist[i] = fminf(dist[i], row[i]);
        __syncthreads();
    }
}

// ---------------------------------------------------------------------------
// In-LDS bitonic sort of 4096 floats (one block per view), ascending.
// ---------------------------------------------------------------------------
__global__ void __launch_bounds__(1024)
sort_kernel(float* __restrict__ w1, float* __restrict__ w2) {
    float* w = blockIdx.x ? w2 : w1;
    __shared__ float s[N];
    const int tid = threadIdx.x;
    for (int i = tid; i < N; i += 1024) s[i] = w[i];
    __syncthreads();
    for (int k = 2; k <= N; k <<= 1) {
        for (int j = k >> 1; j > 0; j >>= 1) {
            for (int p = 0; p < N / 1024; ++p) {
                const int t   = tid + p * 1024;
                const int ixj = t ^ j;
                if (ixj > t) {
                    const float a = s[t], b = s[ixj];
                    const bool asc = ((t & k) == 0);
                    if ((a > b) == asc) { s[t] = b; s[ixj] = a; }
                }
            }
            __syncthreads();
        }
    }
    for (int i = tid; i < N; i += 1024) w[i] = s[i];
}

// ---------------------------------------------------------------------------
// Final scalar: MSE(x1,x2) + sum_k |sorted_d1[k] - sorted_d2[k]|, k < N-1.
// Single block -> fixed reduction order -> deterministic.
// ---------------------------------------------------------------------------
__global__ void __launch_bounds__(1024)
final_kernel(const float* __restrict__ x1, const float* __restrict__ x2,
             const float* __restrict__ s1, const float* __restrict__ s2,
             float* __restrict__ out) {
    __shared__ float red[1024];
    const int tid = threadIdx.x;
    const int TOT = N * K;
    float mse = 0.f;
    for (int i = tid; i < TOT; i += 1024) {
        const float d = x1[i] - x2[i];
        mse += d * d;
    }
    float acc = mse * (1.0f / (float)TOT);
    for (int i = tid; i < N - 1; i += 1024)
        acc += fabsf(s1[i] - s2[i]);
    red[tid] = acc;
    __syncthreads();
    for (int s = 512; s > 0; s >>= 1) {
        if (tid < s) red[tid] += red[tid + s];
        __syncthreads();
    }
    if (tid == 0) out[0] = red[0];
}

// ---------------------------------------------------------------------------
extern "C" void kernel_launch(void* const* d_in, const int* in_sizes, int n_in,
                              void* d_out, int out_size, void* d_ws, size_t ws_size,
                              hipStream_t stream) {
    (void)in_sizes; (void)n_in; (void)out_size; (void)ws_size;
    const float* x1 = (const float*)d_in[0];
    const float* x2 = (const float*)d_in[1];

    float* ws  = (float*)d_ws;
    float* D1  = ws;                         // N*N
    float* D2  = D1 + (size_t)N * N;         // N*N
    float* sq1 = D2 + (size_t)N * N;         // N
    float* sq2 = sq1 + N;                    // N
    float* w1  = sq2 + N;                    // N (N-1 weights + inf pad)
    float* w2  = w1 + N;                     // N

    sqnorm_kernel<<<(2 * N + 255) / 256, 256, 0, stream>>>(x1, x2, sq1, sq2);

    dim3 grid(N / 64, (N / 16) / 4, 2);      // (64, 64, 2), 4 waves/block
    dist_wmma_kernel<<<grid, 128, 0, stream>>>(x1, x2, sq1, sq2, D1, D2);

    prim_kernel<<<2, 256, 0, stream>>>(D1, D2, w1, w2);
    sort_kernel<<<2, 1024, 0, stream>>>(w1, w2);
    final_kernel<<<1, 1024, 0, stream>>>(x1, x2, w1, w2, (float*)d_out);
}